// Transfrmr_bidaf_32280974197404
// MI455X (gfx1250) — compile-verified
//
#include <hip/hip_runtime.h>
#include <math.h>
#include <stdint.h>

#define Bn 64
#define CL 1024
#define QL 128
#define D2 512

typedef __attribute__((ext_vector_type(2))) float v2f;
typedef __attribute__((ext_vector_type(8))) float v8f;

// ---- workspace layout (floats) ----
#define SQ_OFF   0u                         // [B*QL]
#define M_OFF    (SQ_OFF + Bn*QL)           // [B*CL]
#define S_OFF    (M_OFF + Bn*CL)            // [B*CL*QL]  (32 MB)
#define BATT_OFF (S_OFF + (size_t)Bn*CL*QL) // [B*CL]
#define PART_OFF (BATT_OFF + Bn*CL)         // [B*8*D2]
#define Q2C_OFF  (PART_OFF + Bn*8*D2)       // [B*D2]

// ------------------------------------------------------------------
// Kernel 1: s_q[b,j] = q[b,j,:] . w_q + b_q   (one wave per row)
// ------------------------------------------------------------------
__global__ void k_sq(const float* __restrict__ q, const float* __restrict__ w_q,
                     const float* __restrict__ b_q, float* __restrict__ sq) {
  int wave = threadIdx.x >> 5, lane = threadIdx.x & 31;
  int j = blockIdx.x * 8 + wave;           // [0, B*QL)
  int b = j / QL, jj = j % QL;
  const float4* q4 = (const float4*)(q + ((size_t)(b * QL + jj)) * D2);
  const float4* w4 = (const float4*)w_q;
  float acc = 0.f;
#pragma unroll
  for (int i = 0; i < 4; ++i) {
    float4 a = q4[i * 32 + lane];
    float4 w = w4[i * 32 + lane];
    acc += a.x * w.x + a.y * w.y + a.z * w.z + a.w * w.w;
  }
#pragma unroll
  for (int m = 16; m; m >>= 1) acc += __shfl_xor(acc, m, 32);
  if (lane == 0) sq[b * QL + jj] = acc + b_q[0];
}

// ------------------------------------------------------------------
// Kernel 2: s[b,i,j] = sum_k c[i,k]*(w_cq[k]*q[j,k] + w_c[k]) + s_q[j] + b_c + b_cq
// 64x128 tile per WG (4 waves). f32 WMMA 16x16x4, K in 128-wide chunks.
// c tile staged with DOUBLE-BUFFERED async global->LDS DMA (ASYNCcnt path);
// B' tile staged with fused w_cq*q + w_c transform. Also writes row max m.
// ------------------------------------------------------------------
__global__ void __launch_bounds__(128)
k_scores(const float* __restrict__ c, const float* __restrict__ q,
         const float* __restrict__ w_c, const float* __restrict__ w_cq,
         const float* __restrict__ bias_c, const float* __restrict__ bias_cq,
         const float* __restrict__ sq, float* __restrict__ s_out,
         float* __restrict__ m_out) {
  __shared__ __align__(16) float c_lds[2][64][132]; // 67.6 KB (double buffer)
  __shared__ __align__(16) float qb_lds[QL][132];   // 67.6 KB
  __shared__ float sq_lds[QL];

  int tid = threadIdx.x;
  int b = blockIdx.y;
  int row0 = blockIdx.x * 64;
  sq_lds[tid] = sq[b * QL + tid];                  // blockDim == QL == 128
  float bias0 = bias_c[0] + bias_cq[0];

  int lane = tid & 31, wave = tid >> 5;
  int l15 = lane & 15;
  int hk = (lane >> 4) << 1;                       // +0 or +2 in K

  // ---- async c-chunk staging: 64 rows x 128 K, b128 per lane ----
  auto stage_c_async = [&](int buf, int kc) {
#pragma unroll
    for (int it = 0; it < 16; ++it) {
      int idx = tid + it * 128;                    // [0, 2048)
      int r = idx >> 5, k4 = idx & 31;
      unsigned voff = (unsigned)(((b * CL + row0 + r) * D2 + kc + k4 * 4) * 4);
      unsigned lds = (unsigned)(uintptr_t)&c_lds[buf][r][k4 * 4];
      asm volatile("global_load_async_to_lds_b128 %0, %1, %2"
                   :: "v"(lds), "v"(voff), "s"(c) : "memory");
    }
  };
  // ---- synchronous B' staging with fused transform ----
  auto stage_qb = [&](int kc) {
    for (int idx = tid; idx < QL * 32; idx += 128) {
      int j = idx >> 5, k4 = idx & 31;
      float4 v  = *(const float4*)(q + ((size_t)(b * QL + j)) * D2 + kc + k4 * 4);
      float4 wq = *(const float4*)(w_cq + kc + k4 * 4);
      float4 wc = *(const float4*)(w_c + kc + k4 * 4);
      float4 r4;
      r4.x = v.x * wq.x + wc.x; r4.y = v.y * wq.y + wc.y;
      r4.z = v.z * wq.z + wc.z; r4.w = v.w * wq.w + wc.w;
      *(float4*)&qb_lds[j][k4 * 4] = r4;
    }
  };

  v8f acc[8];
  v8f z = {};
#pragma unroll
  for (int t = 0; t < 8; ++t) acc[t] = z;

  // prologue: stage chunk 0
  stage_c_async(0, 0);
  stage_qb(0);
  asm volatile("s_wait_asynccnt 0x0" ::: "memory");
  __syncthreads();

  int cur = 0;
  for (int kci = 0; kci < 4; ++kci) {
    int kc = kci * 128;
    if (kci < 3) stage_c_async(cur ^ 1, kc + 128);  // DMA overlaps WMMA below
#pragma unroll 4
    for (int ks = 0; ks < 32; ++ks) {
      int k0 = ks * 4 + hk;
      v2f a = *(const v2f*)&c_lds[cur][wave * 16 + l15][k0];
#pragma unroll
      for (int t = 0; t < 8; ++t) {
        v2f bf = *(const v2f*)&qb_lds[t * 16 + l15][k0];
        acc[t] = __builtin_amdgcn_wmma_f32_16x16x4_f32(
            false, a, false, bf, (short)0, acc[t], false, false);
      }
    }
    __syncthreads();                                // qb now free to overwrite
    if (kci < 3) {
      stage_qb(kc + 128);
      asm volatile("s_wait_asynccnt 0x0" ::: "memory");
      __syncthreads();
      cur ^= 1;
    }
  }
  // bias add (per-j s_q + scalar biases)
#pragma unroll
  for (int t = 0; t < 8; ++t) {
    float bj = sq_lds[t * 16 + l15] + bias0;
#pragma unroll
    for (int r = 0; r < 8; ++r) acc[t][r] += bj;
  }
  // store s + row max. C layout: vgpr r, lanes<16 -> row r, lanes>=16 -> row r+8.
  int rowbase = row0 + wave * 16 + ((lane >> 4) << 3);
#pragma unroll
  for (int r = 0; r < 8; ++r) {
    float rm = acc[0][r];
#pragma unroll
    for (int t = 1; t < 8; ++t) rm = fmaxf(rm, acc[t][r]);
    rm = fmaxf(rm, __shfl_xor(rm, 1, 32));
    rm = fmaxf(rm, __shfl_xor(rm, 2, 32));
    rm = fmaxf(rm, __shfl_xor(rm, 4, 32));
    rm = fmaxf(rm, __shfl_xor(rm, 8, 32));
    int row = rowbase + r;
    if (l15 == 0) m_out[b * CL + row] = rm;
    float* srow = s_out + ((size_t)(b * CL + row)) * QL;
#pragma unroll
    for (int t = 0; t < 8; ++t) srow[t * 16 + l15] = acc[t][r];
  }
}

// ------------------------------------------------------------------
// Kernel 3: b_att[b,:] = softmax_i(m[b,:])
// ------------------------------------------------------------------
__global__ void k_batt(const float* __restrict__ m, float* __restrict__ batt) {
  __shared__ float red[256];
  int b = blockIdx.x, tid = threadIdx.x;
  const float* mb = m + b * CL;
  float v[4], mx = -INFINITY;
#pragma unroll
  for (int i = 0; i < 4; ++i) { v[i] = mb[tid + i * 256]; mx = fmaxf(mx, v[i]); }
  red[tid] = mx; __syncthreads();
  for (int s = 128; s; s >>= 1) { if (tid < s) red[tid] = fmaxf(red[tid], red[tid + s]); __syncthreads(); }
  mx = red[0]; __syncthreads();
  float e[4], sum = 0.f;
#pragma unroll
  for (int i = 0; i < 4; ++i) { e[i] = __expf(v[i] - mx); sum += e[i]; }
  red[tid] = sum; __syncthreads();
  for (int s = 128; s; s >>= 1) { if (tid < s) red[tid] += red[tid + s]; __syncthreads(); }
  float rs = 1.f / red[0];
#pragma unroll
  for (int i = 0; i < 4; ++i) batt[b * CL + tid + i * 256] = e[i] * rs;
}

// ------------------------------------------------------------------
// Kernel 4a/4b: q2c[b,d] = sum_i b_att[b,i] * c[b,i,d]
// (streaming rows; prefetch 16 rows / 32 KB ahead to keep HBM pipe fed)
// ------------------------------------------------------------------
__global__ void k_q2c_part(const float* __restrict__ c, const float* __restrict__ batt,
                           float* __restrict__ part) {
  int b = blockIdx.y, ci = blockIdx.x, d = threadIdx.x;   // blockDim = 512
  int row0 = ci * 128;
  float acc = 0.f;
  for (int i = 0; i < 128; ++i) {
    __builtin_prefetch(&c[((size_t)(b * CL + row0 + i + 16)) * D2 + d], 0, 0);
    acc += batt[b * CL + row0 + i] * c[((size_t)(b * CL + row0 + i)) * D2 + d];
  }
  part[(size_t)(b * 8 + ci) * D2 + d] = acc;
}

__global__ void k_q2c_reduce(const float* __restrict__ part, float* __restrict__ q2c) {
  int b = blockIdx.x, d = threadIdx.x;
  float acc = 0.f;
#pragma unroll
  for (int ci = 0; ci < 8; ++ci) acc += part[(size_t)(b * 8 + ci) * D2 + d];
  q2c[b * D2 + d] = acc;
}

// ------------------------------------------------------------------
// Kernel 5: a = softmax(s) rowwise; c2q = a @ q (WMMA, q staged in LDS
// transposed so consecutive-K is contiguous); emit x = [c, c2q, c*c2q, c*q2c].
// 32 rows x 512 cols per WG (8 waves: 2 row strips x 4 col ranges),
// d staged in 2 chunks of 256.
// ------------------------------------------------------------------
__global__ void __launch_bounds__(256)
k_out(const float* __restrict__ c, const float* __restrict__ q,
      const float* __restrict__ s_ws, const float* __restrict__ m_ws,
      const float* __restrict__ q2c, float* __restrict__ out) {
  __shared__ __align__(16) float qT[256][132];     // 135 KB: qT[d][j]
  __shared__ __align__(16) float a_lds[32][132];   // 16.9 KB

  int tid = threadIdx.x;
  int b = blockIdx.y;
  int row0 = blockIdx.x * 32;

  // row softmax: row r handled by 8 consecutive lanes (16 cols each)
  {
    int r = tid >> 3, g = tid & 7;
    const float* srow = s_ws + ((size_t)(b * CL + row0 + r)) * QL + g * 16;
    float mrow = m_ws[b * CL + row0 + r];
    float ev[16], sum = 0.f;
#pragma unroll
    for (int jj = 0; jj < 16; ++jj) { ev[jj] = __expf(srow[jj] - mrow); sum += ev[jj]; }
    sum += __shfl_xor(sum, 1, 32);
    sum += __shfl_xor(sum, 2, 32);
    sum += __shfl_xor(sum, 4, 32);
    float rs = 1.f / sum;
#pragma unroll
    for (int jj = 0; jj < 16; ++jj) a_lds[r][g * 16 + jj] = ev[jj] * rs;
  }

  int lane = tid & 31, wave = tid >> 5;
  int strip = wave >> 2;        // 0..1  (16-row strip)
  int colw = wave & 3;          // 0..3  (64-col range within 256-chunk)
  int l15 = lane & 15;
  int hk = (lane >> 4) << 1;
  __syncthreads();

  for (int dc = 0; dc < 2; ++dc) {
    int dbase = dc * 256;
    __syncthreads();
    // stage qT[d][j] for this 256-d chunk (coalesced reads, scatter to LDS)
    for (int idx = tid; idx < QL * 64; idx += 256) {
      int j = idx >> 6, d4 = idx & 63;
      float4 v = *(const float4*)(q + ((size_t)(b * QL + j)) * D2 + dbase + d4 * 4);
      qT[d4 * 4 + 0][j] = v.x; qT[d4 * 4 + 1][j] = v.y;
      qT[d4 * 4 + 2][j] = v.z; qT[d4 * 4 + 3][j] = v.w;
    }
    __syncthreads();

    v8f acc[4];
    v8f z = {};
#pragma unroll
    for (int t = 0; t < 4; ++t) acc[t] = z;
#pragma unroll 4
    for (int ks = 0; ks < 32; ++ks) {
      int k0 = ks * 4 + hk;
      v2f a = *(const v2f*)&a_lds[strip * 16 + l15][k0];
#pragma unroll
      for (int t = 0; t < 4; ++t) {
        v2f bf = *(const v2f*)&qT[colw * 64 + t * 16 + l15][k0];
        acc[t] = __builtin_amdgcn_wmma_f32_16x16x4_f32(
            false, a, false, bf, (short)0, acc[t], false, false);
      }
    }
    // emit outputs for this d chunk
    int rowb = row0 + strip * 16 + ((lane >> 4) << 3);
#pragma unroll
    for (int t = 0; t < 4; ++t) {
      int d = dbase + colw * 64 + t * 16 + l15;
      float q2cd = q2c[b * D2 + d];
#pragma unroll
      for (int r = 0; r < 8; ++r) {
        int row = rowb + r;
        float cv = c[((size_t)(b * CL + row)) * D2 + d];
        float c2qv = acc[t][r];
        float* o = out + ((size_t)(b * CL + row)) * (4 * D2);
        o[d] = cv;
        o[512 + d] = c2qv;
        o[1024 + d] = cv * c2qv;
        o[1536 + d] = cv * q2cd;
      }
    }
  }
}

// ------------------------------------------------------------------
extern "C" void kernel_launch(void* const* d_in, const int* in_sizes, int n_in,
                              void* d_out, int out_size, void* d_ws, size_t ws_size,
                              hipStream_t stream) {
  const float* c    = (const float*)d_in[0];
  const float* q    = (const float*)d_in[1];
  const float* w_c  = (const float*)d_in[2];
  const float* b_c  = (const float*)d_in[3];
  const float* w_q  = (const float*)d_in[4];
  const float* b_q  = (const float*)d_in[5];
  const float* w_cq = (const float*)d_in[6];
  const float* b_cq = (const float*)d_in[7];
  float* out = (float*)d_out;
  float* ws  = (float*)d_ws;

  float* sq   = ws + SQ_OFF;
  float* m    = ws + M_OFF;
  float* s    = ws + S_OFF;
  float* batt = ws + BATT_OFF;
  float* part = ws + PART_OFF;
  float* q2c  = ws + Q2C_OFF;

  k_sq<<<dim3(Bn * QL / 8), dim3(256), 0, stream>>>(q, w_q, b_q, sq);
  k_scores<<<dim3(CL / 64, Bn), dim3(128), 0, stream>>>(c, q, w_c, w_cq, b_c, b_cq,
                                                        sq, s, m);
  k_batt<<<dim3(Bn), dim3(256), 0, stream>>>(m, batt);
  k_q2c_part<<<dim3(8, Bn), dim3(512), 0, stream>>>(c, batt, part);
  k_q2c_reduce<<<dim3(Bn), dim3(512), 0, stream>>>(part, q2c);
  k_out<<<dim3(CL / 32, Bn), dim3(256), 0, stream>>>(c, q, s, m, q2c, out);
}